// layers_88622355186375
// MI455X (gfx1250) — compile-verified
//
#include <hip/hip_runtime.h>

#define N_NODES 30000
#define NNZ_E   480000

typedef __attribute__((ext_vector_type(2))) float v2f;
typedef __attribute__((ext_vector_type(8))) float v8f;

// ---------------------------------------------------------------- zero fill
__global__ __launch_bounds__(256) void zero_f32(float* __restrict__ p, int n) {
  int i = blockIdx.x * 256 + threadIdx.x;
  if (i < n) p[i] = 0.0f;
}

// --------------------------------------------- layer-1 SpMM (both directions)
// lK[rows[e]] += vals[e] * x0[cols[e]]   (A  @ x0)
// rK[cols[e]] += vals[e] * x1[rows[e]]   (A^T @ x1)
// D = 64; 16 threads per edge, float4 per thread.
__global__ __launch_bounds__(256) void spmm_dual64(
    const int* __restrict__ rows, const int* __restrict__ cols,
    const float* __restrict__ vals,
    const float* __restrict__ x0, const float* __restrict__ x1,
    float* __restrict__ lK, float* __restrict__ rK) {
  int idx = blockIdx.x * 256 + threadIdx.x;
  int e = idx >> 4;
  if (e >= NNZ_E) return;
  int d = (idx & 15) << 2;
  int r = rows[e], c = cols[e];
  float v = vals[e];
  const float4 f0 = *reinterpret_cast<const float4*>(x0 + (size_t)c * 64 + d);
  const float4 f1 = *reinterpret_cast<const float4*>(x1 + (size_t)r * 64 + d);
  float* pl = lK + (size_t)r * 64 + d;
  float* pr = rK + (size_t)c * 64 + d;
  atomicAdd(pl + 0, v * f0.x);
  atomicAdd(pl + 1, v * f0.y);
  atomicAdd(pl + 2, v * f0.z);
  atomicAdd(pl + 3, v * f0.w);
  atomicAdd(pr + 0, v * f1.x);
  atomicAdd(pr + 1, v * f1.y);
  atomicAdd(pr + 2, v * f1.z);
  atomicAdd(pr + 3, v * f1.w);
}

// ------------------------------------------------- layer-2 SpMM (one direction)
// lK2[rows[e]] += vals[e] * x0[cols[e]] ; D = 256; 64 threads per edge.
__global__ __launch_bounds__(256) void spmm_single256(
    const int* __restrict__ rows, const int* __restrict__ cols,
    const float* __restrict__ vals,
    const float* __restrict__ x0, float* __restrict__ lK2) {
  int idx = blockIdx.x * 256 + threadIdx.x;
  int e = idx >> 6;
  if (e >= NNZ_E) return;
  int d = (idx & 63) << 2;
  int r = rows[e], c = cols[e];
  float v = vals[e];
  const float4 f = *reinterpret_cast<const float4*>(x0 + (size_t)c * 256 + d);
  float* pl = lK2 + (size_t)r * 256 + d;
  atomicAdd(pl + 0, v * f.x);
  atomicAdd(pl + 1, v * f.y);
  atomicAdd(pl + 2, v * f.z);
  atomicAdd(pl + 3, v * f.w);
}

// ------------------------------------------------------------- fused dual GEMM
// out = relu(A @ W1 + (A .* Xm) @ W2 + b1 + b2)
// A, Xm: [n, KD] row-major; W1, W2: [KD, ND] row-major; out: [n, ND].
// One wave computes one 16x16 output tile via V_WMMA_F32_16X16X4_F32,
// sharing a single f32 accumulator for both GEMMs (sum is associative).
// 4 waves per block tile the N dimension.
template <int KD, int ND>
__global__ __launch_bounds__(128) void dual_gemm_bias_relu(
    const float* __restrict__ A, const float* __restrict__ Xm,
    const float* __restrict__ W1, const float* __restrict__ W2,
    const float* __restrict__ b1, const float* __restrict__ b2,
    float* __restrict__ out) {
  const int lane = threadIdx.x & 31;
  const int wid  = threadIdx.x >> 5;
  const int m0 = blockIdx.y * 16;
  const int n0 = (blockIdx.x * 4 + wid) * 16;
  const int half = lane >> 4;     // 0: lanes 0-15, 1: lanes 16-31
  const int l15  = lane & 15;
  const int arow = m0 + l15;      // A-matrix row this lane supplies
  const int bcol = n0 + l15;      // B-matrix / output column this lane supplies

  const float* Ar = A  + (size_t)arow * KD;
  const float* Mr = Xm + (size_t)arow * KD;

  v8f acc = {};
#pragma unroll
  for (int k0 = 0; k0 < KD; k0 += 4) {
    const int ka = k0 + half * 2; // A layout: lanes 0-15 -> K=0,1 ; lanes 16-31 -> K=2,3
    v2f a1 = *reinterpret_cast<const v2f*>(Ar + ka);
    v2f xm = *reinterpret_cast<const v2f*>(Mr + ka);
    v2f a2 = a1 * xm;             // elementwise (A .* Xm)
    v2f bw1, bw2;                 // B layout symmetric: N striped over lanes per VGPR
    bw1.x = W1[(size_t)(ka + 0) * ND + bcol];
    bw1.y = W1[(size_t)(ka + 1) * ND + bcol];
    bw2.x = W2[(size_t)(ka + 0) * ND + bcol];
    bw2.y = W2[(size_t)(ka + 1) * ND + bcol];
    // 8 args: (neg_a, A, neg_b, B, c_mod, C, reuse_a, reuse_b)
    acc = __builtin_amdgcn_wmma_f32_16x16x4_f32(false, a1, false, bw1,
                                                (short)0, acc, false, false);
    acc = __builtin_amdgcn_wmma_f32_16x16x4_f32(false, a2, false, bw2,
                                                (short)0, acc, false, false);
  }

  const float bias = b1[bcol] + b2[bcol];
#pragma unroll
  for (int j = 0; j < 8; ++j) {   // C/D: VGPR j -> M = m0 + j + 8*half, N = bcol
    const int row = m0 + j + half * 8;
    const float vv = acc[j] + bias;
    out[(size_t)row * ND + bcol] = fmaxf(vv, 0.0f);
  }
}

// ------------------------------------------------------------------ launcher
extern "C" void kernel_launch(void* const* d_in, const int* in_sizes, int n_in,
                              void* d_out, int out_size, void* d_ws, size_t ws_size,
                              hipStream_t stream) {
  const float* l_feat = (const float*)d_in[0];
  const float* r_feat = (const float*)d_in[1];
  const int*   rows   = (const int*)d_in[2];
  const int*   cols   = (const int*)d_in[3];
  const float* vals   = (const float*)d_in[4];
  const float* W1 = (const float*)d_in[5];
  const float* b1 = (const float*)d_in[6];
  const float* W2 = (const float*)d_in[7];
  const float* b2 = (const float*)d_in[8];
  const float* W3 = (const float*)d_in[9];
  const float* b3 = (const float*)d_in[10];
  const float* W4 = (const float*)d_in[11];
  const float* b4 = (const float*)d_in[12];
  float* out = (float*)d_out;

  // workspace layout (floats): lK[N*64] rK[N*64] y1[N*256] z1[N*256] lK2[N*256]
  float* ws  = (float*)d_ws;
  float* lK  = ws;
  float* rK  = lK + (size_t)N_NODES * 64;
  float* y1  = rK + (size_t)N_NODES * 64;
  float* z1  = y1 + (size_t)N_NODES * 256;
  float* lK2 = z1 + (size_t)N_NODES * 256;

  // ---- layer 1 ----
  {
    int n = N_NODES * 128;  // lK + rK contiguous
    zero_f32<<<(n + 255) / 256, 256, 0, stream>>>(lK, n);
  }
  spmm_dual64<<<(NNZ_E * 16) / 256, 256, 0, stream>>>(rows, cols, vals,
                                                      r_feat, l_feat, lK, rK);
  dim3 g1(256 / 64, N_NODES / 16);
  dual_gemm_bias_relu<64, 256><<<g1, 128, 0, stream>>>(lK, l_feat, W1, W2, b1, b2, y1);
  dual_gemm_bias_relu<64, 256><<<g1, 128, 0, stream>>>(rK, r_feat, W1, W2, b1, b2, z1);

  // ---- layer 2 (only the "l" branch of the second inception is used) ----
  {
    int n = N_NODES * 256;
    zero_f32<<<(n + 255) / 256, 256, 0, stream>>>(lK2, n);
  }
  spmm_single256<<<(NNZ_E * 64) / 256, 256, 0, stream>>>(rows, cols, vals, z1, lK2);
  dim3 g2(128 / 64, N_NODES / 16);
  dual_gemm_bias_relu<256, 128><<<g2, 128, 0, stream>>>(lK2, y1, W3, W4, b3, b4, out);
}